// DWT_22213570855414
// MI455X (gfx1250) — compile-verified
//
#include <hip/hip_runtime.h>

typedef __attribute__((ext_vector_type(2))) float v2f;
typedef __attribute__((ext_vector_type(8))) float v8f;

#define B_N   8
#define H_N   512
#define W_N   512
#define C_N   16
#define OL    259            // (512 + 8 - 1) / 2
#define HO_T  8              // output rows (H) per tile
#define WO_T  16             // output cols (W) per tile
#define TW    42             // padded LDS row length (need 40; 42 spreads banks)
#define TROWS (HO_T * C_N)   // 128 intermediate rows (ho, c)
#define NIN   (2 * HO_T + 6) // 22 input H samples per column

__device__ __forceinline__ int reflect_idx(int i, int n) {
  // pywt 'symmetric': -1 -> 0, -2 -> 1 ; n -> n-1, n+1 -> n-2 (single
  // reflection suffices: pad <= 39 << n)
  i = (i < 0) ? (-i - 1) : i;
  i = (i >= n) ? (2 * n - 1 - i) : i;
  return i;
}

__global__ __launch_bounds__(256) void dwt2d_db4_fused(
    const float* __restrict__ x, const float* __restrict__ dec_lo,
    float* __restrict__ out) {
  __shared__ float t_lds[2][TROWS][TW]; // a/d H-DWT tiles: 2*128*42*4 = 43008 B
  __shared__ float fpad[2][72];         // zero-padded banded filters (taps at [30..37])

  const int tid = threadIdx.x;
  const int b   = blockIdx.z;
  const int ho0 = blockIdx.y * HO_T;
  const int wo0 = blockIdx.x * WO_T;

  // warm L2/L0 for this tile's first input line (emits global_prefetch_b8)
  {
    const int w0 = reflect_idx(2 * wo0 - 6 + (tid >> 4), W_N);
    const int h0 = reflect_idx(2 * ho0 - 6, H_N);
    __builtin_prefetch(&x[(((size_t)b * H_N + h0) * W_N + w0) * C_N + (tid & 15)], 0, 0);
  }

  // Effective downsampling-conv filters (pywt): out[k] = sum_m g[m] * x[2k-6+m]
  //   g_lo[m] = dec_lo[7-m]
  //   g_hi[m] = dec_hi[7-m] = (m odd ? -1 : +1) * dec_lo[m]
  float glo[8], ghi[8];
#pragma unroll
  for (int m = 0; m < 8; ++m) {
    glo[m] = dec_lo[7 - m];
    ghi[m] = (m & 1) ? -dec_lo[m] : dec_lo[m];
  }

  // Zero the padded filter bands, then drop the 8 taps at [30..37].
  for (int i = tid; i < 2 * 72; i += 256) (&fpad[0][0])[i] = 0.0f;
  __syncthreads();
  if (tid < 8) {
    fpad[0][30 + tid] = dec_lo[7 - tid];
    fpad[1][30 + tid] = (tid & 1) ? -dec_lo[tid] : dec_lo[tid];
  }

  // ---------------- Stage 1: DWT along H -> LDS (transposed for stage 2) ----
  // Column (wt, c): t index s=wt corresponds to global w = 2*wo0 - 6 + wt.
  // lane -> (c fast, wt slow): 32 lanes cover 128 contiguous bytes per load.
#pragma unroll 1
  for (int p = 0; p < 3; ++p) {
    const int wt = p * 16 + (tid >> 4);
    const int c  = tid & 15;
    if (wt < 40) {
      const int w = reflect_idx(2 * wo0 - 6 + wt, W_N);
      float xv[NIN];
#pragma unroll
      for (int i = 0; i < NIN; ++i) {
        const int h = reflect_idx(2 * ho0 - 6 + i, H_N);
        xv[i] = x[(((size_t)b * H_N + h) * W_N + w) * C_N + c];
      }
#pragma unroll
      for (int ho = 0; ho < HO_T; ++ho) {
        float a = 0.0f, d = 0.0f;
#pragma unroll
        for (int m = 0; m < 8; ++m) {
          const float v = xv[2 * ho + m];
          a = fmaf(glo[m], v, a);
          d = fmaf(ghi[m], v, d);
        }
        t_lds[0][ho * C_N + c][wt] = a;
        t_lds[1][ho * C_N + c][wt] = d;
      }
    }
  }
  __syncthreads();

  const size_t plane = (size_t)B_N * OL * OL * C_N;

  // ---------------- Stage 2: DWT along W as 16x16x4 f32 WMMA GEMM ----------
  // D[wo, r] = sum_s A[wo, s] * B[s, r],  A[wo, s] = g[s - 2*wo] (banded),
  // B[s, r] = t[src][r][s], r = ho*16 + c.  K=40 split into 10 chunks of 4.
#if __has_builtin(__builtin_amdgcn_wmma_f32_16x16x4_f32)
  const int lane = tid & 31;
  // wave id is uniform per wave: force it scalar so T/combo/jg/oh live in
  // SGPRs and the oh/interior checks become scalar branches.
  const int wave = __builtin_amdgcn_readfirstlane(tid >> 5);
  const int half = lane >> 4; // 0: lanes 0-15, 1: lanes 16-31
  const int n    = lane & 15; // A row (M=wo) and B column (N=r%16 -> c)
  const bool interior = (wo0 + WO_T <= OL); // uniform: 16 of 17 W-tiles

#pragma unroll 1
  for (int T = wave; T < HO_T * 4; T += 8) { // uniform per wave: EXEC all-ones
    const int combo = T & 3;      // 0:aa 1:ad 2:da 3:dd (fixed per wave)
    const int jg    = T >> 2;     // output row (ho) within tile
    const int src   = combo >> 1; // 0: a-tile, 1: d-tile
    const int filt  = combo & 1;  // 0: lo, 1: hi
    const float* trow = &t_lds[src][jg * C_N + n][0];
    const float* frow = &fpad[filt][30 - 2 * n]; // index stays >= 0 (n<=15)
    v8f acc = {};
#pragma unroll
    for (int kc = 0; kc < 10; ++kc) {
      const int s0 = 4 * kc + 2 * half;            // even -> 8B-aligned b64 loads
      const v2f aval = *(const v2f*)(frow + s0);   // A: v0=K(2*half), v1=K+1
      const v2f bval = *(const v2f*)(trow + s0);   // B: v0=K(2*half), v1=K+1
      acc = __builtin_amdgcn_wmma_f32_16x16x4_f32(
          false, aval, false, bval, (short)0, acc, false, false);
    }
    // D layout: VGPR v, lane -> out[M = v + 8*half][N = n] = out[wo][ho=jg, c=n]
    const int oh = ho0 + jg; // scalar
    if (oh < OL) {
      float* obase = out + (size_t)combo * plane +
                     (((size_t)b * OL + oh) * OL) * (size_t)C_N + n;
      if (interior) {
        // fast path: straight-line non-temporal stores (output never re-read;
        // keep L2 for the reused input instead)
#pragma unroll
        for (int v = 0; v < 8; ++v)
          __builtin_nontemporal_store(acc[v],
              &obase[(size_t)(wo0 + v + 8 * half) * C_N]);
      } else {
#pragma unroll
        for (int v = 0; v < 8; ++v) {
          const int wo = wo0 + v + 8 * half;
          if (wo < OL)
            __builtin_nontemporal_store(acc[v], &obase[(size_t)wo * C_N]);
        }
      }
    }
  }
#else
  // Scalar fallback (keeps the file compiling if the f32 WMMA builtin is absent)
  for (int idx = tid; idx < HO_T * WO_T * C_N * 4; idx += 256) {
    const int c = idx & 15;
    int r = idx >> 4;
    const int wo_l = r & 15; r >>= 4;
    const int combo = r & 3;
    const int jg = r >> 2;
    const int src = combo >> 1, filt = combo & 1;
    float acc = 0.0f;
#pragma unroll
    for (int m = 0; m < 8; ++m) {
      const float g = filt ? ghi[m] : glo[m];
      acc = fmaf(g, t_lds[src][jg * C_N + c][2 * wo_l + m], acc);
    }
    const int oh = ho0 + jg, ow = wo0 + wo_l;
    if (oh < OL && ow < OL)
      out[(size_t)combo * plane + (((size_t)b * OL + oh) * OL + ow) * C_N + c] = acc;
  }
#endif
}

extern "C" void kernel_launch(void* const* d_in, const int* in_sizes, int n_in,
                              void* d_out, int out_size, void* d_ws, size_t ws_size,
                              hipStream_t stream) {
  (void)in_sizes; (void)n_in; (void)out_size; (void)d_ws; (void)ws_size;
  const float* x      = (const float*)d_in[0];
  const float* dec_lo = (const float*)d_in[1];
  float* out          = (float*)d_out;
  dim3 grid((OL + WO_T - 1) / WO_T, (OL + HO_T - 1) / HO_T, B_N); // (17, 33, 8)
  dwt2d_db4_fused<<<grid, dim3(256, 1, 1), 0, stream>>>(x, dec_lo, out);
}